// Warper_57062935495226
// MI455X (gfx1250) — compile-verified
//
#include <hip/hip_runtime.h>

typedef float v2f __attribute__((ext_vector_type(2)));
typedef float v8f __attribute__((ext_vector_type(8)));

#define OUT_N 16
#define OUT_C 3
#define OUT_H 512
#define OUT_W 512
#define T36   36

#define LN2F 0.69314718055994530942f

// TPS radial basis: U = d^2 * ln(d + 1e-6) = (d^2 * ln2) * log2(d + 1e-6)
// Raw HW sqrt/log: inputs bounded in [1e-6, ~sqrt(2)], no denorm/inf fixups needed.
__device__ __forceinline__ float tps_u(float gx, float gy, float2 c) {
    float dx = gx - c.x;
    float dy = gy - c.y;
    float d2 = dx * dx + dy * dy;
    float d  = __builtin_amdgcn_sqrtf(d2);
    return (d2 * LN2F) * __builtin_amdgcn_logf(d + 1e-6f);
}

__global__ __launch_bounds__(256) void tps_warp_wmma_kernel(
    const float* __restrict__ theta,   // (16, 38, 2)
    const float* __restrict__ mask,    // (16, 3, 512, 512)
    float* __restrict__ out)           // (16, 3, 512, 512)
{
    __shared__ float  s_w[T36][OUT_N][2];   // effective TPS weights w_eff[t][n][c]
    __shared__ float  s_a[OUT_N][3][2];     // affine rows a[n][k][c]
    __shared__ float2 s_ctrl[T36];          // control-point coords (cx, cy)
    __shared__ float2 s_z[8][16][16];       // per-wave [pixel][batch] -> (zx, zy)

    const int tid  = threadIdx.x;
    const int lane = tid & 31;
    const int wave = tid >> 5;

    // ---- Phase 0a: control-point table (6x6 grid on [0,1]^2) ----
    if (tid >= 64 && tid < 64 + T36) {
        int t  = tid - 64;
        int ti = t / 6;
        int tj = t - ti * 6;
        float2 c;
        c.x = (float)tj * 0.2f;
        c.y = (float)ti * 0.2f;
        s_ctrl[t] = c;
    }

    // ---- Phase 0b: build w_eff = concat(-sum(w), w) and affine rows in LDS ----
    if (tid < 32) {
        const int n = tid >> 1;
        const int c = tid & 1;
        const float* tn = theta + n * 76 + c;   // theta[n, r, c], stride 2 over r
        float s = 0.0f;
        #pragma unroll 5
        for (int r = 0; r < 35; ++r) {
            float v = tn[r * 2];
            s += v;
            s_w[r + 1][n][c] = v;
        }
        s_w[0][n][c] = -s;
        s_a[n][0][c] = tn[35 * 2];
        s_a[n][1][c] = tn[36 * 2];
        s_a[n][2][c] = tn[37 * 2];
    }
    __syncthreads();

    // ---- Phase 1: WMMA contraction  b[pixel][n][c] = sum_t U[pixel][t] * w_eff[t][n][c]
    // A (16x4 f32): lanes 0-15 hold M=lane, K={kb, kb+1}; lanes 16-31 M=lane-16, K={kb+2, kb+3}
    // B (4x16 f32): lanes 0-15 hold N=lane, K={kb, kb+1}; lanes 16-31 N=lane-16, K={kb+2, kb+3}
    const int   h        = blockIdx.y;
    const int   col_base = blockIdx.x * 128 + wave * 16;
    const float gy       = (float)h * (1.0f / 511.0f);

    const int   mn    = lane & 15;              // pixel index for A, batch index for B
    const int   khalf = (lane >> 4) << 1;       // +0 for lanes 0-15, +2 for lanes 16-31
    const float gx_a  = (float)(col_base + mn) * (1.0f / 511.0f);

    v8f accx = {};   // zx over [pixel 16][batch 16]
    v8f accy = {};   // zy

    #pragma unroll
    for (int kk = 0; kk < 9; ++kk) {            // K = 36 = 9 steps of 4
        const int kb = kk * 4 + khalf;
        float2 c0 = s_ctrl[kb];
        float2 c1 = s_ctrl[kb + 1];
        v2f a;
        a.x = tps_u(gx_a, gy, c0);
        a.y = tps_u(gx_a, gy, c1);
        v2f bx, by;
        bx.x = s_w[kb][mn][0];
        bx.y = s_w[kb + 1][mn][0];
        by.x = s_w[kb][mn][1];
        by.y = s_w[kb + 1][mn][1];
        accx = __builtin_amdgcn_wmma_f32_16x16x4_f32(false, a, false, bx, (short)0, accx, false, false);
        accy = __builtin_amdgcn_wmma_f32_16x16x4_f32(false, a, false, by, (short)0, accy, false, false);
    }

    // D layout: VGPR r, lanes 0-15 -> (pixel=r,   n=lane); lanes 16-31 -> (pixel=r+8, n=lane-16)
    {
        const int prow = (lane >> 4) << 3;
        const int n    = lane & 15;
        #pragma unroll
        for (int r = 0; r < 8; ++r) {
            float2 z;
            z.x = accx[r];
            z.y = accy[r];
            s_z[wave][prow + r][n] = z;
        }
    }
    __syncthreads();

    // ---- Phase 2: bilinear grid-sample; lane owns pixel (lane&15), batches (lane>>4)+2j ----
    const int   p     = lane & 15;
    const int   w_idx = col_base + p;
    const float gx    = (float)w_idx * (1.0f / 511.0f);

    #pragma unroll 2
    for (int j = 0; j < 8; ++j) {
        const int n = (lane >> 4) + j * 2;
        float2 z = s_z[wave][p][n];
        float zx = z.x + s_a[n][0][0] + s_a[n][1][0] * gx + s_a[n][2][0] * gy;
        float zy = z.y + s_a[n][0][1] + s_a[n][1][1] * gx + s_a[n][2][1] * gy;

        // grid = (loc + z)*2 - 1; x = (grid+1)*W/2 - 0.5 = (loc+z)*W - 0.5
        float x = (gx + zx) * (float)OUT_W - 0.5f;
        float y = (gy + zy) * (float)OUT_H - 0.5f;

        float x0f = floorf(x);
        float y0f = floorf(y);
        float tx = x - x0f;
        float ty = y - y0f;
        int ix0 = (int)x0f;
        int iy0 = (int)y0f;
        int ix1 = ix0 + 1;
        int iy1 = iy0 + 1;

        float vx0 = (ix0 >= 0 && ix0 < OUT_W) ? 1.0f : 0.0f;
        float vx1 = (ix1 >= 0 && ix1 < OUT_W) ? 1.0f : 0.0f;
        float vy0 = (iy0 >= 0 && iy0 < OUT_H) ? 1.0f : 0.0f;
        float vy1 = (iy1 >= 0 && iy1 < OUT_H) ? 1.0f : 0.0f;

        float w00 = (1.0f - tx) * (1.0f - ty) * vx0 * vy0;
        float w10 = tx * (1.0f - ty) * vx1 * vy0;
        float w01 = (1.0f - tx) * ty * vx0 * vy1;
        float w11 = tx * ty * vx1 * vy1;

        int cx0 = min(max(ix0, 0), OUT_W - 1);
        int cx1 = min(max(ix1, 0), OUT_W - 1);
        int cy0 = min(max(iy0, 0), OUT_H - 1);
        int cy1 = min(max(iy1, 0), OUT_H - 1);

        const float* mb = mask + (size_t)n * OUT_C * OUT_H * OUT_W;
        #pragma unroll
        for (int c = 0; c < OUT_C; ++c) {
            const float* mc = mb + (size_t)c * OUT_H * OUT_W;
            float v = mc[cy0 * OUT_W + cx0] * w00
                    + mc[cy0 * OUT_W + cx1] * w10
                    + mc[cy1 * OUT_W + cx0] * w01
                    + mc[cy1 * OUT_W + cx1] * w11;
            out[(((size_t)n * OUT_C + c) * OUT_H + h) * OUT_W + w_idx] = v;
        }
    }
}

extern "C" void kernel_launch(void* const* d_in, const int* in_sizes, int n_in,
                              void* d_out, int out_size, void* d_ws, size_t ws_size,
                              hipStream_t stream) {
    (void)in_sizes; (void)n_in; (void)out_size; (void)d_ws; (void)ws_size;
    const float* theta = (const float*)d_in[0];   // (16, 38, 2) f32
    const float* mask  = (const float*)d_in[1];   // (16, 3, 512, 512) f32
    float* out = (float*)d_out;                   // (16, 3, 512, 512) f32

    dim3 grid(OUT_W / 128, OUT_H);   // block covers 128 cols of one row; 8 waves x 16 px
    dim3 block(256);
    tps_warp_wmma_kernel<<<grid, block, 0, stream>>>(theta, mask, out);
}